// Decoder_66391604462295
// MI455X (gfx1250) — compile-verified
//
#include <hip/hip_runtime.h>
#include <math.h>

// Problem constants (match reference)
#define Bsz   16
#define Nn    256
#define Ld    64
#define Hh    32
#define HE    128
#define EO    64
#define NMPS  3
#define ALPHA 0.2f

typedef __attribute__((ext_vector_type(2))) float v2f;
typedef __attribute__((ext_vector_type(8))) float v8f;

__device__ __forceinline__ float lrelu(float v) { return (v >= 0.f) ? v : ALPHA * v; }

// ---------------------------------------------------------------------------
// h0 = ((x @ W_lin + b_lin).reshape(B,N,L)) @ W_in + b_in   -> [B,N,32]
// one block per (b,n), 64 threads
// ---------------------------------------------------------------------------
__global__ __launch_bounds__(64) void k_input(const float* __restrict__ x,
                                              const float* __restrict__ W_lin,
                                              const float* __restrict__ b_lin,
                                              const float* __restrict__ W_in,
                                              const float* __restrict__ b_in,
                                              float* __restrict__ h) {
  const int b = blockIdx.x >> 8;
  const int n = blockIdx.x & 255;
  const int c = threadIdx.x;                 // 0..63
  __shared__ float xs[Ld];
  __shared__ float ts[Ld];
  xs[c] = x[b * Ld + c];
  __syncthreads();
  float acc = b_lin[n * Ld + c];
#pragma unroll 8
  for (int l = 0; l < Ld; ++l) acc += xs[l] * W_lin[l * (Nn * Ld) + n * Ld + c];
  ts[c] = acc;
  __syncthreads();
  if (c < Hh) {
    float a2 = b_in[c];
#pragma unroll 8
    for (int l = 0; l < Ld; ++l) a2 += ts[l] * W_in[l * Hh + c];
    h[(b * Nn + n) * Hh + c] = a2;
  }
}

// ---------------------------------------------------------------------------
// rcv = h @ We1[t][:H],  snd = h @ We1[t][H:]   (biases added later)
// one block per (b,n), 128 threads (thread = output k)
// ---------------------------------------------------------------------------
__global__ __launch_bounds__(128) void k_rcv_snd(const float* __restrict__ h,
                                                 const float* __restrict__ We1,
                                                 float* __restrict__ rcv,
                                                 float* __restrict__ snd, int t) {
  const int bn = blockIdx.x;
  const int k = threadIdx.x;                 // 0..127
  __shared__ float hs[Hh];
  if (k < Hh) hs[k] = h[bn * Hh + k];
  __syncthreads();
  const float* W1 = We1 + (size_t)t * (2 * Hh) * HE;
  float a1 = 0.f, a2 = 0.f;
#pragma unroll 8
  for (int hh = 0; hh < Hh; ++hh) {
    const float hv = hs[hh];
    a1 += hv * W1[hh * HE + k];
    a2 += hv * W1[(Hh + hh) * HE + k];
  }
  rcv[bn * HE + k] = a1;
  snd[bn * HE + k] = a2;
}

// ---------------------------------------------------------------------------
// Edge MLP + sum-aggregate (dominant GEMM), fp32 WMMA 16x16x4.
//   m[b,i,:] = sum_j lrelu( lrelu(rcv_i + snd_j + be1) @ We2 + be2 )
// block = (b,i), 128 threads = 4 waves; wave w owns output columns
// [w*16, w*16+16). Loops 16 sender tiles of 16 rows.
// WMMA layouts used (per ISA 7.12.2):
//   A 16x4 f32: lane&15 = row M, (vgpr, lane>>4) -> K = 4kk + (lane>>4)*2 + vgpr
//   B 4x16 f32: lane&15 = col N, same (vgpr, lane-half)->K map as A
//   C 16x16:    lane&15 = col N, vgpr r -> row r + (lane>>4)*8
// Row mapping of C is irrelevant: all 16 rows are summed.
// ---------------------------------------------------------------------------
#define EPAD 132   // LDS row stride (floats): 16 rows * 132 -> conflict-free b64 reads

__global__ __launch_bounds__(128) void k_edge(const float* __restrict__ rcv,
                                              const float* __restrict__ snd,
                                              const float* __restrict__ be1,
                                              const float* __restrict__ We2,
                                              const float* __restrict__ be2,
                                              float* __restrict__ m, int t) {
  const int b    = blockIdx.x >> 8;
  const int i    = blockIdx.x & 255;
  const int tid  = threadIdx.x;
  const int wave = tid >> 5;                 // 0..3 -> N-tile
  const int lane = tid & 31;
  const int lh   = lane >> 4;                // lane half
  const int ln   = lane & 15;

  __shared__ float Etile[16 * EPAD];

  // receiver contribution + edge bias for this thread's k (= tid)
  const float rc = rcv[(b * Nn + i) * HE + tid] + be1[t * HE + tid];

  // Preload this wave's B fragments: columns nc = wave*16 + ln, all 32 K-steps
  const int nc = wave * 16 + ln;
  const float* W2 = We2 + (size_t)t * HE * EO;
  v2f Bf[32];
#pragma unroll
  for (int kk = 0; kk < 32; ++kk) {
    const int k0 = kk * 4 + lh * 2;
    Bf[kk].x = W2[(k0    ) * EO + nc];
    Bf[kk].y = W2[(k0 + 1) * EO + nc];
  }
  const float bias2 = be2[t * EO + nc];

  float macc[8];
#pragma unroll
  for (int r = 0; r < 8; ++r) macc[r] = 0.f;

  const float* sndb = snd + (size_t)(b * Nn) * HE;

  for (int jt = 0; jt < Nn / 16; ++jt) {
    __syncthreads();                         // protect Etile vs. prev reads
    // Build E tile: 16 sender rows x 128 k; thread handles column k = tid
#pragma unroll 4
    for (int jr = 0; jr < 16; ++jr) {
      Etile[jr * EPAD + tid] = lrelu(rc + sndb[(jt * 16 + jr) * HE + tid]);
    }
    __syncthreads();

    v8f C = {};
#pragma unroll
    for (int kk = 0; kk < 32; ++kk) {
      const int k0 = kk * 4 + lh * 2;
      v2f A;
      A.x = Etile[ln * EPAD + k0];
      A.y = Etile[ln * EPAD + k0 + 1];
      C = __builtin_amdgcn_wmma_f32_16x16x4_f32(
          /*neg_a=*/false, A, /*neg_b=*/false, Bf[kk],
          /*c_mod=*/(short)0, C, /*reuse_a=*/false, /*reuse_b=*/false);
    }
    // e = lrelu(C + be2); accumulate 8 rows per lane-half into macc
#pragma unroll
    for (int r = 0; r < 8; ++r) macc[r] += lrelu(C[r] + bias2);
  }

  // reduce over this lane-half's 8 rows, then across halves (other 8 rows)
  float colsum = 0.f;
#pragma unroll
  for (int r = 0; r < 8; ++r) colsum += macc[r];
  colsum += __shfl_xor(colsum, 16, 32);
  if (lane < 16) m[(b * Nn + i) * EO + nc] = colsum;
}

// ---------------------------------------------------------------------------
// Node MLP: n = concat(h, m); h' = lrelu(lrelu(n@Wn0+bn0)@Wn1+bn1)
// optionally tanh into d_out on the final round. One block per (b,n), 32 thr.
// ---------------------------------------------------------------------------
__global__ __launch_bounds__(32) void k_node(const float* __restrict__ h,
                                             const float* __restrict__ m,
                                             const float* __restrict__ Wn0,
                                             const float* __restrict__ bn0,
                                             const float* __restrict__ Wn1,
                                             const float* __restrict__ bn1,
                                             float* __restrict__ hout, int t,
                                             int apply_tanh) {
  const int bn = blockIdx.x;
  const int tid = threadIdx.x;               // 0..31
  __shared__ float nf[Hh + EO];
  __shared__ float t0[Hh];
  nf[tid]          = h[bn * Hh + tid];
  nf[Hh + tid]     = m[bn * EO + tid];
  nf[Hh + 32 + tid]= m[bn * EO + 32 + tid];
  __syncthreads();
  const float* W0 = Wn0 + (size_t)t * (Hh + EO) * Hh;
  float acc = bn0[t * Hh + tid];
#pragma unroll 8
  for (int f = 0; f < Hh + EO; ++f) acc += nf[f] * W0[f * Hh + tid];
  t0[tid] = lrelu(acc);
  __syncthreads();
  const float* W1 = Wn1 + (size_t)t * Hh * Hh;
  float a2 = bn1[t * Hh + tid];
#pragma unroll 8
  for (int f = 0; f < Hh; ++f) a2 += t0[f] * W1[f * Hh + tid];
  a2 = lrelu(a2);
  hout[bn * Hh + tid] = apply_tanh ? tanhf(a2) : a2;
}

// ---------------------------------------------------------------------------
extern "C" void kernel_launch(void* const* d_in, const int* in_sizes, int n_in,
                              void* d_out, int out_size, void* d_ws, size_t ws_size,
                              hipStream_t stream) {
  const float* x     = (const float*)d_in[0];
  const float* W_lin = (const float*)d_in[1];
  const float* b_lin = (const float*)d_in[2];
  const float* W_in  = (const float*)d_in[3];
  const float* b_in  = (const float*)d_in[4];
  const float* We1   = (const float*)d_in[5];
  const float* be1   = (const float*)d_in[6];
  const float* We2   = (const float*)d_in[7];
  const float* be2   = (const float*)d_in[8];
  const float* Wn0   = (const float*)d_in[9];
  const float* bn0   = (const float*)d_in[10];
  const float* Wn1   = (const float*)d_in[11];
  const float* bn1   = (const float*)d_in[12];
  float* out = (float*)d_out;

  // Workspace layout (floats): h | rcv | snd | m  = ~5.5 MB total
  float* ws  = (float*)d_ws;
  float* h   = ws;                                   // B*N*H   = 131072
  float* rcv = h   + Bsz * Nn * Hh;                  // B*N*HE  = 524288
  float* snd = rcv + Bsz * Nn * HE;                  // B*N*HE  = 524288
  float* mbuf= snd + Bsz * Nn * HE;                  // B*N*EO  = 262144

  const int BN = Bsz * Nn;                           // 4096

  k_input<<<BN, 64, 0, stream>>>(x, W_lin, b_lin, W_in, b_in, h);
  for (int t = 0; t < NMPS; ++t) {
    k_rcv_snd<<<BN, 128, 0, stream>>>(h, We1, rcv, snd, t);
    k_edge<<<BN, 128, 0, stream>>>(rcv, snd, be1, We2, be2, mbuf, t);
    k_node<<<BN, 32, 0, stream>>>(h, mbuf, Wn0, bn0, Wn1, bn1,
                                  (t == NMPS - 1) ? out : h, t, t == NMPS - 1);
  }
}